// HLFMLP_74852690035114
// MI455X (gfx1250) — compile-verified
//
#include <hip/hip_runtime.h>
#include <cstdint>
#include <cstddef>

// Problem constants (from reference): B=8, C=128, H=W=256
#define NIMG   1024          // B*C
#define HP     128           // pooled H
#define WP     128           // pooled W
#define WF     65            // rfft width
#define TILES  520           // (128*65)/16 position tiles per (b,k)
#define LAMBD  0.01f

typedef __attribute__((ext_vector_type(2))) float v2f;
typedef __attribute__((ext_vector_type(8))) float v8f;

// ============ CDNA5 async global<->LDS DMA (ASYNCcnt-tracked) ============
__device__ __forceinline__ unsigned lds_off(const void* p) {
  // LDS aperture: addr[31:0] is the wave-relative LDS byte offset
  return (unsigned)(uintptr_t)p;
}
__device__ __forceinline__ void async_g2l_b32(unsigned lds, const void* g) {
  asm volatile("global_load_async_to_lds_b32 %0, %1, off"
               :: "v"(lds), "v"((unsigned long long)(uintptr_t)g) : "memory");
}
__device__ __forceinline__ void async_g2l_b64(unsigned lds, const void* g) {
  asm volatile("global_load_async_to_lds_b64 %0, %1, off"
               :: "v"(lds), "v"((unsigned long long)(uintptr_t)g) : "memory");
}
__device__ __forceinline__ void async_l2g_b64(const void* g, unsigned lds) {
  asm volatile("global_store_async_from_lds_b64 %0, %1, off"
               :: "v"((unsigned long long)(uintptr_t)g), "v"(lds) : "memory");
}
__device__ __forceinline__ void wait_async0() {
  asm volatile("s_wait_asynccnt 0x0" ::: "memory");
}
__device__ __forceinline__ void wait_ds0() {
  asm volatile("s_wait_dscnt 0x0" ::: "memory");
}

// ======================= 1) pool: max2x2 + avg2x2 =======================
__global__ void pool_kernel(const float* __restrict__ x, float* __restrict__ y) {
  int idx = blockIdx.x * blockDim.x + threadIdx.x;   // 1024*128*128 exact
  int w  = idx & 127;
  int h  = (idx >> 7) & 127;
  int bc = idx >> 14;
  const float* p = x + (((size_t)bc * 256 + 2 * h) * 256 + 2 * w);
  float a = p[0], b = p[1], c = p[256], d = p[257];
  float mx = fmaxf(fmaxf(a, b), fmaxf(c, d));
  y[idx] = mx + 0.25f * (a + b + c + d);
}

// ======================= radix-2 FFT (serial per thread, data in LDS) ====
__device__ __forceinline__ void fill_tw(float2* tw, int t, int nthr) {
  for (int j = t; j < 64; j += nthr) {
    float th = 6.283185307179586f * (float)j / 128.0f;
    tw[j] = make_float2(__cosf(th), __sinf(th));
  }
}

// in-place 128-pt complex FFT; stride in float2 units; sgn=+1 fwd, -1 inv (no 1/N)
__device__ __forceinline__ void fft128(float2* d, int stride, const float2* tw, float sgn) {
#pragma unroll 1
  for (int i = 1; i < 128; ++i) {
    int j = (int)(__brev((unsigned)i) >> 25);
    if (j > i) {
      float2 a = d[i * stride], b = d[j * stride];
      d[i * stride] = b; d[j * stride] = a;
    }
  }
#pragma unroll 1
  for (int s = 0; s < 7; ++s) {
    int half = 1 << s;
    int tsh  = 6 - s;
#pragma unroll 1
    for (int k = 0; k < 128; k += (half << 1)) {
      for (int j = 0; j < half; ++j) {
        float2 t = tw[j << tsh];
        float wr = t.x, wi = -sgn * t.y;
        float2* pu = d + (size_t)(k + j) * stride;
        float2* pv = d + (size_t)(k + j + half) * stride;
        float2 u = *pu, v = *pv;
        float tr = wr * v.x - wi * v.y;
        float ti = wr * v.y + wi * v.x;
        *pu = make_float2(u.x + tr, u.y + ti);
        *pv = make_float2(u.x - tr, u.y - ti);
      }
    }
  }
}

// ======================= 2) row rfft: y -> F[h][0..64] (unnormalized) ====
__global__ void rowfft_kernel(const float* __restrict__ y, float2* __restrict__ F) {
  __shared__ float2 buf[32][129];   // pad row pitch 129 -> conflict-free
  __shared__ float2 tw[64];
  int t = threadIdx.x;              // 32 threads, one row each
  fill_tw(&tw[0], t, 32);
  size_t base = (size_t)blockIdx.x * (32 * WP);
  // zero imag slots via DS, DMA the reals straight into the .x slots
  for (int i = t; i < 32 * WP; i += 32) buf[i >> 7][i & 127].y = 0.0f;
  for (int i = t; i < 32 * WP; i += 32)
    async_g2l_b32(lds_off(&buf[i >> 7][i & 127].x), y + base + i);
  wait_async0();
  __syncthreads();
  fft128(&buf[t][0], 1, tw, 1.0f);
  wait_ds0();                       // FFT results visible in LDS before DMA-out
  __syncthreads();
  size_t obase = (size_t)blockIdx.x * (32 * WF);
  for (int i = t; i < 32 * WF; i += 32)
    async_l2g_b64(F + obase + i, lds_off(&buf[i / WF][i % WF]));
  // async stores drain at s_endpgm (implicit wait-idle)
}

// ======================= 3/5) column FFT over all 1024*65 columns ========
__global__ void colfft_kernel(float2* __restrict__ F, float sgn) {
  __shared__ float2 buf[128 * 33];  // tile pitch 33 -> conflict-free
  __shared__ float2 tw[64];
  int t = threadIdx.x;
  fill_tw(tw, t, 32);
  int cg  = blockIdx.x * 32 + t;    // 2080*32 = 66560 columns exact
  int img = cg / WF;
  int c   = cg - img * WF;
  float2* base = F + (size_t)img * (HP * WF) + c;
#pragma unroll 1
  for (int h = 0; h < 128; ++h)
    async_g2l_b64(lds_off(&buf[h * 33 + t]), base + (size_t)h * WF);
  wait_async0();
  __syncthreads();
  fft128(&buf[t], 33, tw, sgn);
  wait_ds0();
  __syncthreads();
#pragma unroll 1
  for (int h = 0; h < 128; ++h)
    async_l2g_b64(base + (size_t)h * WF, lds_off(&buf[h * 33 + t]));
}

// ======================= 4) spectral gating via fp32 WMMA ================
__device__ __forceinline__ v8f wmma4(v2f a, v2f b, v8f c) {
  // V_WMMA_F32_16X16X4_F32  (A 16x4, B 4x16, C/D 16x16)
  return __builtin_amdgcn_wmma_f32_16x16x4_f32(false, a, false, b, (short)0, c,
                                               false, false);
}
__device__ __forceinline__ float sshrink(float v) {
  return v > LAMBD ? v - LAMBD : (v < -LAMBD ? v + LAMBD : 0.0f);
}

__global__ void __launch_bounds__(256)
gate_kernel(float2* __restrict__ F,
            const float* __restrict__ w1, const float* __restrict__ b1,
            const float* __restrict__ w2, const float* __restrict__ b2) {
  // one wave = one 16-position tile of one (b,k) block; 8 waves/block
  __shared__ float lr[8][16][20];   // layer-1 output (real), [wave][M][N], pitch 20
  __shared__ float li[8][16][20];   // layer-1 output (imag)
  const int wv   = threadIdx.x >> 5;
  const int lane = threadIdx.x & 31;
  const int gw   = blockIdx.x * 8 + wv;   // 0..33279 exact
  const int tile = gw % TILES;
  const int bk   = gw / TILES;            // b*8 + k
  const int k    = bk & 7;
  const int b    = bk >> 3;
  const int m    = lane & 15;             // A-layout row / C-layout column
  const int hi   = lane >> 4;
  const int koff = hi * 2;
  const float FNORM = 1.0f / 128.0f;      // folded ortho fwd-FFT normalization

  const int p = tile * 16 + m;            // frequency position (h*65+w)
  const int h = p / WF;
  const int w = p - h * WF;
  const size_t chanStride = (size_t)HP * WF;
  const size_t blkBase = ((size_t)b * 128 + (size_t)k * 16) * chanStride;
  const size_t posOff  = (size_t)h * WF + w;

  // --- A fragments: f at (row m, channels koff+{0,1} per K-step kk) ---
  v2f Ar[4], Ai[4];
#pragma unroll
  for (int kk = 0; kk < 4; ++kk) {
    float2 f0 = F[blkBase + (size_t)(4 * kk + koff)     * chanStride + posOff];
    float2 f1 = F[blkBase + (size_t)(4 * kk + koff + 1) * chanStride + posOff];
    Ar[kk] = (v2f){f0.x * FNORM, f1.x * FNORM};
    Ai[kk] = (v2f){f0.y * FNORM, f1.y * FNORM};
  }

  // --- B fragments: weights W[k][K][N], N = m; pre-negate Wi for the
  //     real-part subtraction (F32 WMMA NEG supports CNeg only) ---
  v2f B1r[4], B1i[4], B1n[4], B2r[4], B2i[4], B2n[4];
#pragma unroll
  for (int kk = 0; kk < 4; ++kk) {
    int K0 = 4 * kk + koff;
    int o  = k * 256 + K0 * 16 + m;       // [part][k][K][N], part stride 2048
    B1r[kk] = (v2f){w1[o],        w1[o + 16]};
    B1i[kk] = (v2f){w1[2048 + o], w1[2048 + o + 16]};
    B1n[kk] = -B1i[kk];
    B2r[kk] = (v2f){w2[o],        w2[o + 16]};
    B2i[kk] = (v2f){w2[2048 + o], w2[2048 + o + 16]};
    B2n[kk] = -B2i[kk];
  }
  float br1 = b1[k * 16 + m], bi1 = b1[128 + k * 16 + m];
  float br2 = b2[k * 16 + m], bi2 = b2[128 + k * 16 + m];

  // --- layer 1: o1r = Xr*Wr - Xi*Wi + br ; o1i = Xi*Wr + Xr*Wi + bi ---
  v8f cr = {br1, br1, br1, br1, br1, br1, br1, br1};
  v8f ci = {bi1, bi1, bi1, bi1, bi1, bi1, bi1, bi1};
#pragma unroll
  for (int kk = 0; kk < 4; ++kk) cr = wmma4(Ar[kk], B1r[kk], cr);
#pragma unroll
  for (int kk = 0; kk < 4; ++kk) cr = wmma4(Ai[kk], B1n[kk], cr);
#pragma unroll
  for (int kk = 0; kk < 4; ++kk) ci = wmma4(Ai[kk], B1r[kk], ci);
#pragma unroll
  for (int kk = 0; kk < 4; ++kk) ci = wmma4(Ar[kk], B1i[kk], ci);

  // --- C-layout -> A-layout redistribution through per-wave LDS tile ---
#pragma unroll
  for (int v = 0; v < 8; ++v) {
    lr[wv][v + 8 * hi][m] = cr[v];
    li[wv][v + 8 * hi][m] = ci[v];
  }
  __builtin_amdgcn_wave_barrier();
  v2f A2r[4], A2i[4];
#pragma unroll
  for (int kk = 0; kk < 4; ++kk) {
    int K0 = 4 * kk + koff;
    A2r[kk] = (v2f){lr[wv][m][K0], lr[wv][m][K0 + 1]};
    A2i[kk] = (v2f){li[wv][m][K0], li[wv][m][K0 + 1]};
  }

  // --- layer 2 ---
  v8f dr = {br2, br2, br2, br2, br2, br2, br2, br2};
  v8f di = {bi2, bi2, bi2, bi2, bi2, bi2, bi2, bi2};
#pragma unroll
  for (int kk = 0; kk < 4; ++kk) dr = wmma4(A2r[kk], B2r[kk], dr);
#pragma unroll
  for (int kk = 0; kk < 4; ++kk) dr = wmma4(A2i[kk], B2n[kk], dr);
#pragma unroll
  for (int kk = 0; kk < 4; ++kk) di = wmma4(A2i[kk], B2r[kk], di);
#pragma unroll
  for (int kk = 0; kk < 4; ++kk) di = wmma4(A2r[kk], B2i[kk], di);

  // --- softshrink + spectral gating z = g*f (in place; tile owned by wave)
  //     f left unnormalized here; the missing 1/128 is folded into rowifft
  const size_t cBase = blkBase + (size_t)m * chanStride;  // channel N = m
#pragma unroll
  for (int v = 0; v < 8; ++v) {
    float gr = sshrink(dr[v]);
    float gi = sshrink(di[v]);
    int M  = v + 8 * hi;
    int p2 = tile * 16 + M;
    int h2 = p2 / WF;
    int w2p = p2 - h2 * WF;
    size_t off = cBase + (size_t)h2 * WF + w2p;
    float2 f = F[off];
    F[off] = make_float2(gr * f.x - gi * f.y, gr * f.y + gi * f.x);
  }
}

// ======================= 6) row inverse rfft + residual ==================
__global__ void rowifft_kernel(const float2* __restrict__ Z, float* __restrict__ y) {
  __shared__ float2 buf[32][129];
  __shared__ float2 tw[64];
  int t = threadIdx.x;
  fill_tw(&tw[0], t, 32);
  size_t zbase = (size_t)blockIdx.x * (32 * WF);
  for (int i = t; i < 32 * WF; i += 32)
    async_g2l_b64(lds_off(&buf[i / WF][i % WF]), Z + zbase + i);
  wait_async0();
  __syncthreads();
#pragma unroll 1
  for (int c = 1; c < 64; ++c) {          // Hermitian extension of own row
    float2 v = buf[t][c];
    buf[t][128 - c] = make_float2(v.x, -v.y);
  }
  fft128(&buf[t][0], 1, tw, -1.0f);
  __syncthreads();
  size_t ybase = (size_t)blockIdx.x * (32 * WP);
  // (1/128 fwd-ortho folded out of colfft) * (1/128 inv-ortho) = 1/16384
  const float sc = 1.0f / 16384.0f;
  for (int i = t; i < 32 * WP; i += 32)
    y[ybase + i] += buf[i >> 7][i & 127].x * sc;
}

// ======================= 7) bilinear x2 upsample + residual ==============
__global__ void upsample_kernel(const float* __restrict__ yg,
                                const float* __restrict__ x,
                                float* __restrict__ out) {
  int idx = blockIdx.x * blockDim.x + threadIdx.x;   // 1024*256*256 exact
  int j  = idx & 255;
  int i  = (idx >> 8) & 255;
  int bc = idx >> 16;
  int h0 = (i >> 1) - 1 + (i & 1);
  int w0 = (j >> 1) - 1 + (j & 1);
  float fh = (i & 1) ? 0.25f : 0.75f;
  float fw = (j & 1) ? 0.25f : 0.75f;
  int h0c = h0 < 0 ? 0 : h0;
  int h1c = h0 + 1 > 127 ? 127 : h0 + 1;
  int w0c = w0 < 0 ? 0 : w0;
  int w1c = w0 + 1 > 127 ? 127 : w0 + 1;
  const float* p = yg + (size_t)bc * (HP * WP);
  float v00 = p[h0c * WP + w0c];
  float v01 = p[h0c * WP + w1c];
  float v10 = p[h1c * WP + w0c];
  float v11 = p[h1c * WP + w1c];
  float top = v00 + fw * (v01 - v00);
  float bot = v10 + fw * (v11 - v10);
  out[idx] = top + fh * (bot - top) + x[idx];
}

// ======================= host launch =====================================
extern "C" void kernel_launch(void* const* d_in, const int* in_sizes, int n_in,
                              void* d_out, int out_size, void* d_ws, size_t ws_size,
                              hipStream_t stream) {
  const float* x  = (const float*)d_in[0];
  const float* w1 = (const float*)d_in[1];
  const float* b1 = (const float*)d_in[2];
  const float* w2 = (const float*)d_in[3];
  const float* b2 = (const float*)d_in[4];
  float* out = (float*)d_out;

  // workspace: y (1024*128*128 f32 = 64 MiB) then F (1024*128*65 cplx = 65 MiB)
  float*  y = (float*)d_ws;
  float2* F = (float2*)((char*)d_ws + (size_t)NIMG * HP * WP * sizeof(float));

  pool_kernel   <<<65536, 256, 0, stream>>>(x, y);
  rowfft_kernel <<<(NIMG * HP) / 32, 32, 0, stream>>>(y, F);
  colfft_kernel <<<(NIMG * WF) / 32, 32, 0, stream>>>(F, 1.0f);
  gate_kernel   <<<4160, 256, 0, stream>>>(F, w1, b1, w2, b2);
  colfft_kernel <<<(NIMG * WF) / 32, 32, 0, stream>>>(F, -1.0f);
  rowifft_kernel<<<(NIMG * HP) / 32, 32, 0, stream>>>(F, y);
  upsample_kernel<<<262144, 256, 0, stream>>>(y, x, out);
}